// GraphConvolutionBSSVD_89240830476776
// MI455X (gfx1250) — compile-verified
//
#include <hip/hip_runtime.h>
#include <hip/hip_bf16.h>

// ---------------------------------------------------------------------------
// GraphConvolution with low-rank (SVD) weight on MI455X (gfx1250).
//   support = ((X @ U) * s) @ Vt          dense, WMMA f32 16x16x4 path
//   out     = relu(segment_sum(support[col]*val, row) + bias)   COO SpMM
// Memory-bound overall; GEMM stage register-blocked (4 N-tiles x 2 M-tiles
// per wave) so B fragments (U, Vt) are shared and A fragments hoisted,
// cutting vector loads per WMMA from ~3 to ~1.25.
// ---------------------------------------------------------------------------

#define IN_F   512
#define OUT_F  512
#define RANK   64
#define MT     2            // row-tiles (16 rows each) per wave

typedef __attribute__((ext_vector_type(2))) float v2f;
typedef __attribute__((ext_vector_type(8))) float v8f;

// LDS row stride for a 16x64 T tile, padded so stage-2 column-pair reads
// across 16 rows hit distinct banks (stride 68 -> 4*row mod 64).
#define TSTRIDE (RANK + 4)

// ---------------------------------------------------------------------------
// Kernel 1: zero the output accumulator (float4 vectorized).
// ---------------------------------------------------------------------------
__global__ __launch_bounds__(256) void zero_f4_kernel(float4* __restrict__ p,
                                                      long long n4) {
    long long gid = (long long)blockIdx.x * blockDim.x + threadIdx.x;
    if (gid < n4) p[gid] = make_float4(0.f, 0.f, 0.f, 0.f);
}

// ---------------------------------------------------------------------------
// Kernel 2: fused low-rank projection, register-blocked.
// One wave handles MT=2 row-tiles (32 rows). 4 waves per block.
// Stage 1: T[mt][16x64] = X_tile @ U, scaled by s   (all 4 N-tiles live)
// Stage 2: support_tile = T[mt] @ Vt                (B shared across mt)
// Fragment layouts (f32 16x16x4):
//   A (16x4): lane<16 -> M=lane, K={0,1}; lane>=16 -> M=lane-16, K={2,3}
//   B (4x16): lane<16 -> N=lane, K={0,1}; lane>=16 -> N=lane-16, K={2,3}
//   C/D     : VGPR r -> M=r (lanes 0-15) / M=r+8 (lanes 16-31), N=lane&15
// ---------------------------------------------------------------------------
__global__ __launch_bounds__(128) void lowrank_gemm_kernel(
    const float* __restrict__ X, const float* __restrict__ U,
    const float* __restrict__ s, const float* __restrict__ Vt,
    float* __restrict__ support, int n_jobs) {
    __shared__ float Tlds[4][MT][16 * TSTRIDE];

    const int lane = threadIdx.x & 31;
    const int wave = threadIdx.x >> 5;
    const int job  = blockIdx.x * 4 + wave;
    if (job >= n_jobs) return;          // wave-uniform: EXEC all-ones below

    const int m0   = job * (16 * MT);
    const int row  = lane & 15;          // M (A-frag) or N (B-frag) index
    const int kb   = (lane >> 4) * 2;    // K base for this half-wave
    const int mrow = (lane >> 4) * 8;    // C/D row base for this half-wave

    const float* Xr[MT];
    #pragma unroll
    for (int mt = 0; mt < MT; ++mt)
        Xr[mt] = X + (size_t)(m0 + mt * 16 + row) * IN_F;

    // ---- Stage 1: T[mt] = (X_tile[mt] @ U) * s ----------------------------
    {
        v8f acc[RANK / 16][MT];
        #pragma unroll
        for (int nt = 0; nt < RANK / 16; ++nt)
            #pragma unroll
            for (int mt = 0; mt < MT; ++mt)
                acc[nt][mt] = (v8f){};

        for (int kk = 0; kk < IN_F; kk += 4) {
            v2f a[MT];
            #pragma unroll
            for (int mt = 0; mt < MT; ++mt) {
                a[mt].x = Xr[mt][kk + kb];
                a[mt].y = Xr[mt][kk + kb + 1];
            }
            #pragma unroll
            for (int nt = 0; nt < RANK / 16; ++nt) {
                v2f b;
                b.x = U[(size_t)(kk + kb) * RANK + nt * 16 + row];
                b.y = U[(size_t)(kk + kb + 1) * RANK + nt * 16 + row];
                #pragma unroll
                for (int mt = 0; mt < MT; ++mt)
                    acc[nt][mt] = __builtin_amdgcn_wmma_f32_16x16x4_f32(
                        false, a[mt], false, b, (short)0, acc[nt][mt],
                        false, false);
            }
        }
        #pragma unroll
        for (int nt = 0; nt < RANK / 16; ++nt) {
            const float sv = s[nt * 16 + row];   // this lane's N column scale
            #pragma unroll
            for (int mt = 0; mt < MT; ++mt)
                #pragma unroll
                for (int r = 0; r < 8; ++r)
                    Tlds[wave][mt][(mrow + r) * TSTRIDE + nt * 16 + row] =
                        acc[nt][mt][r] * sv;
        }
    }
    // Per-wave-private LDS slabs: the compiler inserts s_wait_dscnt for the
    // RAW dependency; no cross-wave barrier needed.

    // ---- Stage 2: support_tile[mt] = T[mt] @ Vt ---------------------------
    for (int nt = 0; nt < OUT_F / 16; ++nt) {
        const int n0 = nt * 16;
        v8f c[MT];
        #pragma unroll
        for (int mt = 0; mt < MT; ++mt) c[mt] = (v8f){};

        #pragma unroll
        for (int kk = 0; kk < RANK; kk += 4) {
            v2f b;
            b.x = Vt[(size_t)(kk + kb) * OUT_F + n0 + row];
            b.y = Vt[(size_t)(kk + kb + 1) * OUT_F + n0 + row];
            #pragma unroll
            for (int mt = 0; mt < MT; ++mt) {
                v2f a;
                a.x = Tlds[wave][mt][row * TSTRIDE + kk + kb];
                a.y = Tlds[wave][mt][row * TSTRIDE + kk + kb + 1];
                c[mt] = __builtin_amdgcn_wmma_f32_16x16x4_f32(
                    false, a, false, b, (short)0, c[mt], false, false);
            }
        }
        #pragma unroll
        for (int mt = 0; mt < MT; ++mt)
            #pragma unroll
            for (int r = 0; r < 8; ++r)
                support[(size_t)(m0 + mt * 16 + mrow + r) * OUT_F + n0 + row] =
                    c[mt][r];
    }
}

// ---------------------------------------------------------------------------
// Kernel 3: COO SpMM scatter. 128 consecutive threads cover one edge
// (128 float4 chunks of the 512-wide feature row): gather is coalesced and
// (mostly) L2-resident (support ~ 205 MB vs 192 MB L2); scatter uses
// global_atomic_add_f32.
// ---------------------------------------------------------------------------
__global__ __launch_bounds__(256) void spmm_scatter_kernel(
    const float* __restrict__ support, const int* __restrict__ erow,
    const int* __restrict__ ecol, const float* __restrict__ eval,
    float* __restrict__ out, long long n_edges) {
    long long gid = (long long)blockIdx.x * blockDim.x + threadIdx.x;
    long long e = gid >> 7;          // 128 chunk-threads per edge
    if (e >= n_edges) return;
    const int c = (int)(gid & 127);

    const int   r   = erow[e];
    const int   col = ecol[e];
    const float v   = eval[e];

    const float4 g = ((const float4*)(support + (size_t)col * OUT_F))[c];
    float* o = out + (size_t)r * OUT_F + (size_t)c * 4;
    atomicAdd(o + 0, g.x * v);
    atomicAdd(o + 1, g.y * v);
    atomicAdd(o + 2, g.z * v);
    atomicAdd(o + 3, g.w * v);
}

// ---------------------------------------------------------------------------
// Kernel 4: out = relu(out + bias), in place, float4 vectorized.
// ---------------------------------------------------------------------------
__global__ __launch_bounds__(256) void bias_relu_kernel(
    float4* __restrict__ out, const float4* __restrict__ bias4, long long n4) {
    long long gid = (long long)blockIdx.x * blockDim.x + threadIdx.x;
    if (gid >= n4) return;
    const int j4 = (int)(gid & (OUT_F / 4 - 1));   // 128 float4 chunks per row
    float4 o = out[gid];
    float4 b = bias4[j4];
    o.x = fmaxf(o.x + b.x, 0.f);
    o.y = fmaxf(o.y + b.y, 0.f);
    o.z = fmaxf(o.z + b.z, 0.f);
    o.w = fmaxf(o.w + b.w, 0.f);
    out[gid] = o;
}

// ---------------------------------------------------------------------------
extern "C" void kernel_launch(void* const* d_in, const int* in_sizes, int n_in,
                              void* d_out, int out_size, void* d_ws, size_t ws_size,
                              hipStream_t stream) {
    const float* X    = (const float*)d_in[0];
    const float* U    = (const float*)d_in[1];
    const float* s    = (const float*)d_in[2];
    const float* Vt   = (const float*)d_in[3];
    const float* bias = (const float*)d_in[4];
    const int*   erow = (const int*)d_in[5];
    const int*   ecol = (const int*)d_in[6];
    const float* eval = (const float*)d_in[7];

    float* out     = (float*)d_out;
    float* support = (float*)d_ws;   // N_NODES * OUT_F floats (~205 MB)

    const int       n_nodes = in_sizes[0] / IN_F;           // 100000
    const long long n_edges = (long long)in_sizes[5];       // 3.2M
    const long long out_e   = (long long)n_nodes * OUT_F;   // 51.2M floats
    const long long out_e4  = out_e / 4;

    // 1) zero the accumulator (d_out)
    {
        long long blocks = (out_e4 + 255) / 256;
        zero_f4_kernel<<<(unsigned)blocks, 256, 0, stream>>>((float4*)out, out_e4);
    }
    // 2) fused low-rank GEMM -> support (d_ws)
    {
        const int n_row_tiles = n_nodes / 16;               // 6250
        const int n_jobs = (n_row_tiles + MT - 1) / MT;     // 3125 wave-jobs
        const int blocks = (n_jobs + 3) / 4;                // 4 waves/block
        lowrank_gemm_kernel<<<blocks, 128, 0, stream>>>(X, U, s, Vt, support,
                                                        n_jobs);
    }
    // 3) SpMM scatter with f32 atomics into d_out
    {
        long long total  = n_edges * 128;                   // chunk-threads
        long long blocks = (total + 255) / 256;
        spmm_scatter_kernel<<<(unsigned)blocks, 256, 0, stream>>>(
            support, erow, ecol, eval, out, n_edges);
    }
    // 4) bias + ReLU in place
    {
        long long blocks = (out_e4 + 255) / 256;
        bias_relu_kernel<<<(unsigned)blocks, 256, 0, stream>>>(
            (float4*)out, (const float4*)bias, out_e4);
    }
}